// AMCRNet_roiextractor_25228637897332
// MI455X (gfx1250) — compile-verified
//
#include <hip/hip_runtime.h>
#include <hip/hip_bf16.h>

// ---------------- problem constants (match reference) ----------------
#define B_       8
#define N_PER_   16
#define N_ROIS_  128
#define C_SLOW_  1024
#define T_SLOW_  8
#define C_FAST_  256
#define T_FAST_  32
#define HF_      16
#define WF_      16
#define HW_      256            // HF*WF
#define C_CAT_   2048
#define C_OUT_   1024
#define T_POOL_  2
#define C_FEAT_  2048           // C_OUT * T_POOL
#define NCOL_    4096           // B * T_POOL * HW  (GEMM N dim)
#define GN_G_    16
#define GN_CPG_  64             // C_OUT / GN_GROUPS
#define SCALE_   (1.0f / 16.0f)
#define EPS_     1e-5f

typedef __attribute__((ext_vector_type(2))) float v2f;
typedef __attribute__((ext_vector_type(8))) float v8f;

// ---------------------------------------------------------------------
// Stage 1: slow/fast fuse + temporal pool -> X[n][c], n = b*512 + tp*256 + hw
// fast rearrange: fused channel = 1024 + (t%4)*256 + cf, t_outer = t/4,
// then average t_outer over [4*tp, 4*tp+4).
// ---------------------------------------------------------------------
__global__ __launch_bounds__(256) void pool_fuse_kernel(
    const float* __restrict__ slow, const float* __restrict__ fast,
    float* __restrict__ X) {
  size_t idx = (size_t)blockIdx.x * 256 + threadIdx.x;      // NCOL_*C_CAT_ total
  int c  = (int)(idx % C_CAT_);
  int n  = (int)(idx / C_CAT_);
  int hw = n % HW_;
  int tp = (n / HW_) & 1;
  int b  = n / (HW_ * T_POOL_);
  float s;
  if (c < C_SLOW_) {
    // slow[b][c][t][hw], t = 4*tp + i
    const float* p = slow + (((size_t)b * C_SLOW_ + c) * T_SLOW_ + tp * 4) * HW_ + hw;
    s = 0.25f * (p[0] + p[HW_] + p[2 * HW_] + p[3 * HW_]);
  } else {
    int j  = (c - C_SLOW_) >> 8;       // t % 4
    int cf = (c - C_SLOW_) & 255;
    // fast[b][cf][t][hw], t = 16*tp + 4*i + j, i = 0..3
    const float* p = fast + (((size_t)b * C_FAST_ + cf) * T_FAST_ + tp * 16 + j) * HW_ + hw;
    s = 0.25f * (p[0] + p[4 * HW_] + p[8 * HW_] + p[12 * HW_]);
  }
  X[idx] = s;
}

// ---------------------------------------------------------------------
// Stage 2: Y[1024][4096] = W[1024][2048] * X^T  via V_WMMA_F32_16X16X4_F32.
// One wave -> 64x64 output tile (4x4 subtiles of 16x16), K stepped by 4.
// A layout (ISA 7.12.2, 32-bit 16x4): lane m = lane%16, VGPR v + half h holds
// K = 2h+v  => one float2 load of conv_w[row][k0+2h .. +1].
// B assumed mirrored (K = 2h+v, N = lane%16) => float2 load from n-major X.
// ---------------------------------------------------------------------
__global__ __launch_bounds__(128) void gemm_wmma_kernel(
    const float* __restrict__ W,   // [C_OUT_][C_CAT_]
    const float* __restrict__ X,   // [NCOL_][C_CAT_]  (n-major)
    float* __restrict__ Y) {       // [C_OUT_][NCOL_]
  const int lane = threadIdx.x & 31;
  const int wid  = threadIdx.x >> 5;
  const int tile = blockIdx.x * 4 + wid;   // 1024 tiles = 16 (M) x 64 (N)
  const int mt = tile >> 6;                // 0..15
  const int nt = tile & 63;                // 0..63
  const int half = lane >> 4;
  const int l16  = lane & 15;
  const int o0 = mt * 64;
  const int n0 = nt * 64;

  v8f acc[4][4];
#pragma unroll
  for (int i = 0; i < 4; ++i)
#pragma unroll
    for (int j = 0; j < 4; ++j) acc[i][j] = (v8f){};

  const float* Abase = W + (size_t)(o0 + l16) * C_CAT_ + 2 * half;
  const float* Bbase = X + (size_t)(n0 + l16) * C_CAT_ + 2 * half;

  for (int k0 = 0; k0 < C_CAT_; k0 += 4) {
    v2f a[4], b[4];
#pragma unroll
    for (int i = 0; i < 4; ++i)
      a[i] = *(const v2f*)(Abase + (size_t)i * 16 * C_CAT_ + k0);
#pragma unroll
    for (int j = 0; j < 4; ++j)
      b[j] = *(const v2f*)(Bbase + (size_t)j * 16 * C_CAT_ + k0);
#pragma unroll
    for (int i = 0; i < 4; ++i)
#pragma unroll
      for (int j = 0; j < 4; ++j)
        acc[i][j] = __builtin_amdgcn_wmma_f32_16x16x4_f32(
            false, a[i], false, b[j], (short)0, acc[i][j], false, false);
  }

  // D layout: VGPR r, lane -> M = r + 8*half, N = lane%16
#pragma unroll
  for (int i = 0; i < 4; ++i)
#pragma unroll
    for (int j = 0; j < 4; ++j)
#pragma unroll
      for (int r = 0; r < 8; ++r) {
        int m = o0 + i * 16 + r + 8 * half;
        int n = n0 + j * 16 + l16;
        Y[(size_t)m * NCOL_ + n] = acc[i][j][r];
      }
}

// ---------------------------------------------------------------------
// Stage 3a: GroupNorm statistics per (b, g): mean & rsqrt(var+eps)
// over o in [64g,64g+64), n in [512b, 512b+512)  (32768 elements).
// ---------------------------------------------------------------------
__global__ __launch_bounds__(256) void gn_stats_kernel(
    const float* __restrict__ Y, float* __restrict__ stats /* [8*16][2] */) {
  int b = blockIdx.x, g = blockIdx.y;
  int t = threadIdx.x;
  float s = 0.f, s2 = 0.f;
  for (int i = t; i < GN_CPG_ * 512; i += 256) {
    int o = GN_CPG_ * g + (i >> 9);
    int n = 512 * b + (i & 511);
    float v = Y[(size_t)o * NCOL_ + n];
    s += v; s2 += v * v;
  }
  __shared__ float rs[256], rq[256];
  rs[t] = s; rq[t] = s2;
  __syncthreads();
  for (int off = 128; off > 0; off >>= 1) {
    if (t < off) { rs[t] += rs[t + off]; rq[t] += rq[t + off]; }
    __syncthreads();
  }
  if (t == 0) {
    const float inv_n = 1.0f / 32768.0f;
    float mu  = rs[0] * inv_n;
    float var = rq[0] * inv_n - mu * mu;
    stats[(b * GN_G_ + g) * 2 + 0] = mu;
    stats[(b * GN_G_ + g) * 2 + 1] = rsqrtf(var + EPS_);
  }
}

// ---------------------------------------------------------------------
// Stage 3b: normalize + affine + ReLU, write feat[b][(o*2+tp)][hw] to d_out.
// ---------------------------------------------------------------------
__global__ __launch_bounds__(256) void gn_apply_kernel(
    const float* __restrict__ Y, const float* __restrict__ stats,
    const float* __restrict__ gamma, const float* __restrict__ beta,
    float* __restrict__ feat_out) {
  size_t idx = (size_t)blockIdx.x * 256 + threadIdx.x;  // C_OUT_*NCOL_
  int n = (int)(idx % NCOL_);
  int o = (int)(idx / NCOL_);
  int hw = n % HW_;
  int tp = (n / HW_) & 1;
  int b  = n / (HW_ * T_POOL_);
  int g  = o / GN_CPG_;
  float mu  = stats[(b * GN_G_ + g) * 2 + 0];
  float inv = stats[(b * GN_G_ + g) * 2 + 1];
  float v = (Y[idx] - mu) * inv * gamma[o] + beta[o];
  v = fmaxf(v, 0.f);
  feat_out[((size_t)b * C_FEAT_ + (o * T_POOL_ + tp)) * HW_ + hw] = v;
}

// ---------------------------------------------------------------------
// Stage 4: collapsed RoIAlign weights  wbar[n][q] = (1/P) sum_p Wmat[n][p][q]
// built via LDS float atomics. One block (256 threads = 256 output pixels)
// per roi; each thread scatters 4 samples x 4 bilinear corners.
// ---------------------------------------------------------------------
__global__ __launch_bounds__(256) void roiw_kernel(
    const float* __restrict__ rois, float* __restrict__ wbar) {
  int n = blockIdx.x;
  int t = threadIdx.x;           // p index (py*16 + px)
  __shared__ float acc[HW_];
  acc[t] = 0.f;
  __syncthreads();

  float x1 = rois[n * 5 + 1] * SCALE_ - 0.5f;
  float y1 = rois[n * 5 + 2] * SCALE_ - 0.5f;
  float x2 = rois[n * 5 + 3] * SCALE_ - 0.5f;
  float y2 = rois[n * 5 + 4] * SCALE_ - 0.5f;
  float bw = (x2 - x1) * (1.0f / 16.0f);
  float bh = (y2 - y1) * (1.0f / 16.0f);
  int py = t >> 4, px = t & 15;

#pragma unroll
  for (int iy = 0; iy < 2; ++iy) {
#pragma unroll
    for (int ix = 0; ix < 2; ++ix) {
      float Yc = y1 + ((float)py + (iy + 0.5f) * 0.5f) * bh;
      float Xc = x1 + ((float)px + (ix + 0.5f) * 0.5f) * bw;
      float valid = (Yc >= -1.0f && Yc <= 16.0f && Xc >= -1.0f && Xc <= 16.0f) ? 1.0f : 0.0f;
      float y = fmaxf(Yc, 0.0f), x = fmaxf(Xc, 0.0f);
      float y0f = floorf(y), x0f = floorf(x);
      bool ye = (y0f >= 15.0f), xe = (x0f >= 15.0f);
      int y0  = ye ? 15 : (int)y0f;
      int y1i = ye ? 15 : (int)y0f + 1;
      int x0  = xe ? 15 : (int)x0f;
      int x1i = xe ? 15 : (int)x0f + 1;
      float ly = ye ? 0.0f : (y - y0f);
      float lx = xe ? 0.0f : (x - x0f);
      float hy = 1.0f - ly, hx = 1.0f - lx;
      float sc = valid * (1.0f / (4.0f * 256.0f));  // /cnt /P
      atomicAdd(&acc[y0 * 16 + x0],  hy * hx * sc);
      atomicAdd(&acc[y0 * 16 + x1i], hy * lx * sc);
      atomicAdd(&acc[y1i * 16 + x0], ly * hx * sc);
      atomicAdd(&acc[y1i * 16 + x1i], ly * lx * sc);
    }
  }
  __syncthreads();
  wbar[(size_t)n * HW_ + t] = acc[t];
}

// ---------------------------------------------------------------------
// Stage 5: roi_feat[n][c] = sum_q wbar[n][q]*feat[b][c][q],
//          roi_pos [n][c] = sum_q wbar[n][q]*pos[c][q].  Fused.
// ---------------------------------------------------------------------
__global__ __launch_bounds__(256) void roi_apply_kernel(
    const float* __restrict__ feat, const float* __restrict__ pos,
    const float* __restrict__ wbar, const float* __restrict__ rois,
    float* __restrict__ roi_feat, float* __restrict__ roi_pos) {
  int n  = blockIdx.y;
  int cf = blockIdx.x * 256 + threadIdx.x;   // 0..2047
  __shared__ float w[HW_];
  if (threadIdx.x < HW_) w[threadIdx.x] = wbar[(size_t)n * HW_ + threadIdx.x];
  __syncthreads();
  int b = (int)rois[n * 5 + 0];
  const float* frow = feat + ((size_t)b * C_FEAT_ + cf) * HW_;
  const float* prow = pos + (size_t)cf * HW_;
  float sf = 0.f, sp = 0.f;
#pragma unroll 4
  for (int q = 0; q < HW_; ++q) {
    float ww = w[q];
    sf += ww * frow[q];
    sp += ww * prow[q];
  }
  roi_feat[(size_t)n * C_FEAT_ + cf] = sf;
  roi_pos [(size_t)n * C_FEAT_ + cf] = sp;
}

// ---------------------------------------------------------------------
extern "C" void kernel_launch(void* const* d_in, const int* in_sizes, int n_in,
                              void* d_out, int out_size, void* d_ws, size_t ws_size,
                              hipStream_t stream) {
  const float* slow   = (const float*)d_in[0];
  const float* fast   = (const float*)d_in[1];
  const float* rois   = (const float*)d_in[2];
  const float* pos    = (const float*)d_in[3];
  const float* conv_w = (const float*)d_in[4];
  const float* gamma  = (const float*)d_in[5];
  const float* beta   = (const float*)d_in[6];

  float* out = (float*)d_out;
  float* feat_out = out;                                       // [8][2048][256]
  float* roi_feat = out + (size_t)B_ * C_FEAT_ * HW_;          // [128][2048]
  float* roi_pos  = roi_feat + (size_t)N_ROIS_ * C_FEAT_;      // [128][2048]

  char* ws = (char*)d_ws;
  float* X     = (float*)ws;                                   // 4096*2048 f32 = 32 MiB
  float* Y     = (float*)(ws + (size_t)NCOL_ * C_CAT_ * 4);    // 1024*4096 f32 = 16 MiB
  float* wbar  = (float*)(ws + (size_t)NCOL_ * C_CAT_ * 4
                             + (size_t)C_OUT_ * NCOL_ * 4);    // 128*256 f32
  float* stats = wbar + (size_t)N_ROIS_ * HW_;                 // 8*16*2 f32

  // 1) fuse + temporal pool -> X[n][c]
  {
    size_t total = (size_t)NCOL_ * C_CAT_;
    pool_fuse_kernel<<<(int)(total / 256), 256, 0, stream>>>(slow, fast, X);
  }
  // 2) fp32 WMMA GEMM -> Y[o][n]
  gemm_wmma_kernel<<<256, 128, 0, stream>>>(conv_w, X, Y);
  // 3) GroupNorm
  gn_stats_kernel<<<dim3(B_, GN_G_), 256, 0, stream>>>(Y, stats);
  {
    size_t total = (size_t)C_OUT_ * NCOL_;
    gn_apply_kernel<<<(int)(total / 256), 256, 0, stream>>>(Y, stats, gamma, beta, feat_out);
  }
  // 4) collapsed RoIAlign weights
  roiw_kernel<<<N_ROIS_, 256, 0, stream>>>(rois, wbar);
  // 5) roi feature / pos dot products
  roi_apply_kernel<<<dim3(C_FEAT_ / 256, N_ROIS_), 256, 0, stream>>>(
      feat_out, pos, wbar, rois, roi_feat, roi_pos);
}